// GenericTreeLSTMCell_48842368090370
// MI455X (gfx1250) — compile-verified
//
#include <hip/hip_runtime.h>
#include <hip/hip_bf16.h>
#include <stdint.h>

typedef __attribute__((ext_vector_type(2))) float v2f;
typedef __attribute__((ext_vector_type(8))) float v8f;

// Problem sizes (fixed by the reference)
#define NN    1024
#define HH    256
#define LDSW  257             // padded h-tile row stride (floats)
#define CH    64              // B rows staged per chunk
#define NCHUNK (HH * HH / CH) // 1024 chunks over the 65536-deep contraction
#define BROW  136             // padded B row stride (136 % 64 == 8 -> the two
                              // lane-halves (rows k, k+2) read disjoint banks)
#define BTILE (CH * BROW)     // floats per B buffer
#define SMEM_FLOATS (2 * 16 * LDSW + 2 * BTILE)
#define SMEM_BYTES  (SMEM_FLOATS * 4)

__device__ __forceinline__ v8f wmma4(v2f a, v2f b, v8f c) {
  // D(16x16 f32) = A(16x4 f32) * B(4x16 f32) + C
  return __builtin_amdgcn_wmma_f32_16x16x4_f32(false, a, false, b, (short)0, c,
                                               false, false);
}

// ---------------------------------------------------------------------------
// Kernel 1 (templated on B leading dimension so all strides are immediates):
//   IS_F=true , LDB=512: f_aggr  pre-activations -> ws out (1024 x 512)
//   IS_F=false, LDB=768: iou     pre-activations -> ws out (1024 x 768)
// Block: 16 rows x 128 cols (8 waves x 16 cols). B is streamed through LDS
// with double-buffered GLOBAL_LOAD_ASYNC_TO_LDS_B128 (ASYNCcnt).
// ---------------------------------------------------------------------------
template <int LDB, bool IS_F>
__global__ __launch_bounds__(256) void treelstm_pre_kernel(
    const float* __restrict__ x,     // (1024,256)
    const float* __restrict__ nh,    // (1024,2,256)
    const float* __restrict__ B,     // (65536, LDB)  A-tensor reshaped
    const float* __restrict__ U1,    // (LDB,256)
    const float* __restrict__ U2,    // (LDB,256)
    const float* __restrict__ W,     // (256 or 768, 256)
    const float* __restrict__ bias1, // b_fa / b_ia      (LDB)
    const float* __restrict__ bias2, // b_f  / b_iou
    float* __restrict__ out)         // (1024, LDB)
{
  extern __shared__ float smem[];
  float* lds_h1 = smem;
  float* lds_h2 = smem + 16 * LDSW;
  float* ldsB   = smem + 2 * 16 * LDSW;

  const int mbase   = blockIdx.x * 16;
  const int colbase = blockIdx.y * 128;

  // Stage the 16-row h1/h2 tiles (cooperative, coalesced).
  for (int t = threadIdx.x; t < 16 * HH; t += 256) {
    const int rr = t >> 8, cc = t & 255;
    lds_h1[rr * LDSW + cc] = nh[(size_t)(mbase + rr) * 512 + cc];
    lds_h2[rr * LDSW + cc] = nh[(size_t)(mbase + rr) * 512 + 256 + cc];
  }

  const int lane   = threadIdx.x & 31;
  const int wave   = threadIdx.x >> 5;
  const int row    = lane & 15;  // A M-row / B N-col within tile
  const int half   = lane >> 4;
  const int k0     = half * 2;   // K slot in first src VGPR for this lane
  const int k1     = k0 + 1;
  const int colloc = wave * 16 + row;   // column within the 128-col swath
  const int col    = colbase + colloc;  // global output column

  // LDS byte address of the B staging buffers (low 32 bits of generic ptr).
  const unsigned ldsB_base = (unsigned)(uintptr_t)(&ldsB[0]);
  const int tid = threadIdx.x;

  v8f acc0 = {}, acc1 = {}, acc2 = {}, acc3 = {};

  // -------- async stage of one 64x128 B chunk (8 x b128 per thread) -------
  auto stage = [&](int c) {
    const float* gchunk = B + (size_t)c * CH * LDB + colbase;
    const unsigned lbase = ldsB_base + (unsigned)((c & 1) * (BTILE * 4));
#pragma unroll
    for (int k = 0; k < 8; ++k) {
      const int g  = k * 256 + tid;   // 16B granule id within chunk
      const int r  = g >> 5;          // 32 granules per 128-col row
      const int c4 = (g & 31) * 4;
      const float* gp   = gchunk + (size_t)r * LDB + c4;
      const unsigned la = lbase + (unsigned)((r * BROW + c4) * 4);
      asm volatile("global_load_async_to_lds_b128 %0, %1, off"
                   :: "v"(la), "v"(gp) : "memory");
    }
  };

  stage(0);
  __syncthreads();  // also covers the h-tile staging above

  // -------- dominant bilinear term: G[n,(i,j)] = h1[n,i]*h2[n,j] ----------
  for (int c = 0; c < NCHUNK; ++c) {
    if (c + 1 < NCHUNK) {
      stage(c + 1);
      asm volatile("s_wait_asynccnt 0x8" ::: "memory"); // chunk c complete
    } else {
      asm volatile("s_wait_asynccnt 0x0" ::: "memory");
    }
    __syncthreads();

    const float* bb  = ldsB + (c & 1) * BTILE;
    const int    i   = (c * CH) >> 8;    // constant within a chunk (64 | 256)
    const int    jb0 = (c * CH) & 255;
    const float  h1v = lds_h1[row * LDSW + i];
    const float* h2p = &lds_h2[row * LDSW + jb0];

#pragma unroll
    for (int p = 0; p < CH; p += 16) {
      v2f a0 = { h1v * h2p[p +  0 + k0], h1v * h2p[p +  0 + k1] };
      v2f a1 = { h1v * h2p[p +  4 + k0], h1v * h2p[p +  4 + k1] };
      v2f a2 = { h1v * h2p[p +  8 + k0], h1v * h2p[p +  8 + k1] };
      v2f a3 = { h1v * h2p[p + 12 + k0], h1v * h2p[p + 12 + k1] };
      v2f b0 = { bb[(p +  0 + k0) * BROW + colloc], bb[(p +  0 + k1) * BROW + colloc] };
      v2f b1 = { bb[(p +  4 + k0) * BROW + colloc], bb[(p +  4 + k1) * BROW + colloc] };
      v2f b2 = { bb[(p +  8 + k0) * BROW + colloc], bb[(p +  8 + k1) * BROW + colloc] };
      v2f b3 = { bb[(p + 12 + k0) * BROW + colloc], bb[(p + 12 + k1) * BROW + colloc] };
      acc0 = wmma4(a0, b0, acc0);
      acc1 = wmma4(a1, b1, acc1);
      acc2 = wmma4(a2, b2, acc2);
      acc3 = wmma4(a3, b3, acc3);
    }
    __syncthreads();  // buffer (c&1) free for re-staging at iteration c+1
  }

  // -------- h1 @ U1.T  (contract over i, 256 deep) ------------------------
  {
    const float* u1p = U1 + (size_t)col * HH;  // U1.T[i, col] = U1[col, i]
    const float* h1r = &lds_h1[row * LDSW];
#pragma unroll 4
    for (int ib = 0; ib < HH; ib += 16) {
      v2f a0 = { h1r[ib +  0 + k0], h1r[ib +  0 + k1] };
      v2f a1 = { h1r[ib +  4 + k0], h1r[ib +  4 + k1] };
      v2f a2 = { h1r[ib +  8 + k0], h1r[ib +  8 + k1] };
      v2f a3 = { h1r[ib + 12 + k0], h1r[ib + 12 + k1] };
      v2f b0 = { u1p[ib +  0 + k0], u1p[ib +  0 + k1] };
      v2f b1 = { u1p[ib +  4 + k0], u1p[ib +  4 + k1] };
      v2f b2 = { u1p[ib +  8 + k0], u1p[ib +  8 + k1] };
      v2f b3 = { u1p[ib + 12 + k0], u1p[ib + 12 + k1] };
      acc0 = wmma4(a0, b0, acc0);
      acc1 = wmma4(a1, b1, acc1);
      acc2 = wmma4(a2, b2, acc2);
      acc3 = wmma4(a3, b3, acc3);
    }
  }

  // -------- h2 @ U2.T ------------------------------------------------------
  {
    const float* u2p = U2 + (size_t)col * HH;
    const float* h2r = &lds_h2[row * LDSW];
#pragma unroll 4
    for (int ib = 0; ib < HH; ib += 16) {
      v2f a0 = { h2r[ib +  0 + k0], h2r[ib +  0 + k1] };
      v2f a1 = { h2r[ib +  4 + k0], h2r[ib +  4 + k1] };
      v2f a2 = { h2r[ib +  8 + k0], h2r[ib +  8 + k1] };
      v2f a3 = { h2r[ib + 12 + k0], h2r[ib + 12 + k1] };
      v2f b0 = { u2p[ib +  0 + k0], u2p[ib +  0 + k1] };
      v2f b1 = { u2p[ib +  4 + k0], u2p[ib +  4 + k1] };
      v2f b2 = { u2p[ib +  8 + k0], u2p[ib +  8 + k1] };
      v2f b3 = { u2p[ib + 12 + k0], u2p[ib + 12 + k1] };
      acc0 = wmma4(a0, b0, acc0);
      acc1 = wmma4(a1, b1, acc1);
      acc2 = wmma4(a2, b2, acc2);
      acc3 = wmma4(a3, b3, acc3);
    }
  }

  // -------- x @ W.T  (f: tiled f_input once; iou: iou_input appears x2) ---
  {
    const float* xr = x + (size_t)(mbase + row) * HH;
    const int    wc = IS_F ? (col & 255) : col;
    const float* wp = W + (size_t)wc * HH;
    const float  s  = IS_F ? 1.0f : 2.0f;
#pragma unroll 4
    for (int ib = 0; ib < HH; ib += 16) {
      v2f a0 = { s * xr[ib +  0 + k0], s * xr[ib +  0 + k1] };
      v2f a1 = { s * xr[ib +  4 + k0], s * xr[ib +  4 + k1] };
      v2f a2 = { s * xr[ib +  8 + k0], s * xr[ib +  8 + k1] };
      v2f a3 = { s * xr[ib + 12 + k0], s * xr[ib + 12 + k1] };
      v2f b0 = { wp[ib +  0 + k0], wp[ib +  0 + k1] };
      v2f b1 = { wp[ib +  4 + k0], wp[ib +  4 + k1] };
      v2f b2 = { wp[ib +  8 + k0], wp[ib +  8 + k1] };
      v2f b3 = { wp[ib + 12 + k0], wp[ib + 12 + k1] };
      acc0 = wmma4(a0, b0, acc0);
      acc1 = wmma4(a1, b1, acc1);
      acc2 = wmma4(a2, b2, acc2);
      acc3 = wmma4(a3, b3, acc3);
    }
  }

  // -------- reduce accumulators, add per-column bias, store ---------------
  v8f acc = acc0 + acc1;
  acc = acc + acc2;
  acc = acc + acc3;

  const float bias = bias1[col] + bias2[IS_F ? (col & 255) : col];
#pragma unroll
  for (int r = 0; r < 8; ++r) {
    const int mr = mbase + half * 8 + r;  // C/D layout: lanes16-31 hold M=r+8
    out[(size_t)mr * LDB + col] = acc[r] + bias;
  }
}

// ---------------------------------------------------------------------------
// Kernel 2: elementwise epilogue (gates + cell update).
// ---------------------------------------------------------------------------
__device__ __forceinline__ float sigmoidf_(float v) {
  return 1.0f / (1.0f + __expf(-v));
}

__global__ __launch_bounds__(256) void treelstm_epilogue(
    const float* __restrict__ ws_f,   // (1024,512) f_aggr pre-activation
    const float* __restrict__ ws_iou, // (1024,768) iou pre-activation
    const float* __restrict__ nc,     // (1024,2,256)
    float* __restrict__ out)          // h (1024,256) then c (1024,256)
{
  const int idx = blockIdx.x * 256 + threadIdx.x;  // 0 .. N*H-1
  if (idx >= NN * HH) return;
  const int n = idx >> 8, hh = idx & 255;

  const float f0 = sigmoidf_(ws_f[(size_t)n * 512 + hh]);
  const float f1 = sigmoidf_(ws_f[(size_t)n * 512 + 256 + hh]);
  const float caggr = f0 * nc[(size_t)n * 512 + hh] +
                      f1 * nc[(size_t)n * 512 + 256 + hh];

  const float iv = sigmoidf_(ws_iou[(size_t)n * 768 + hh]);
  const float ov = sigmoidf_(ws_iou[(size_t)n * 768 + 256 + hh]);
  const float uv = tanhf(ws_iou[(size_t)n * 768 + 512 + hh]);

  const float c = iv * uv + caggr;
  const float h = ov * tanhf(c);

  out[idx] = h;            // h output
  out[NN * HH + idx] = c;  // c output
}

// ---------------------------------------------------------------------------
extern "C" void kernel_launch(void* const* d_in, const int* in_sizes, int n_in,
                              void* d_out, int out_size, void* d_ws,
                              size_t ws_size, hipStream_t stream) {
  (void)in_sizes; (void)n_in; (void)out_size; (void)ws_size;

  const float* x      = (const float*)d_in[0];
  const float* nh     = (const float*)d_in[1];
  const float* nc     = (const float*)d_in[2];
  const float* W_iou  = (const float*)d_in[3];
  const float* W_f    = (const float*)d_in[4];
  const float* b_iou  = (const float*)d_in[5];
  const float* b_f    = (const float*)d_in[6];
  const float* A_f    = (const float*)d_in[7];
  const float* U1_f   = (const float*)d_in[8];
  const float* U2_f   = (const float*)d_in[9];
  const float* b_fa   = (const float*)d_in[10];
  const float* A_iou  = (const float*)d_in[11];
  const float* U1_iou = (const float*)d_in[12];
  const float* U2_iou = (const float*)d_in[13];
  const float* b_ia   = (const float*)d_in[14];

  float* ws_f   = (float*)d_ws;            // 1024*512 floats (2 MB)
  float* ws_iou = ws_f + (size_t)NN * 512; // 1024*768 floats (3 MB)

  // f branch: 64 m-tiles x 4 column groups of 128
  treelstm_pre_kernel<512, true><<<dim3(NN / 16, 4), 256, SMEM_BYTES, stream>>>(
      x, nh, A_f, U1_f, U2_f, W_f, b_fa, b_f, ws_f);

  // iou branch: 64 m-tiles x 6 column groups of 128
  treelstm_pre_kernel<768, false><<<dim3(NN / 16, 6), 256, SMEM_BYTES, stream>>>(
      x, nh, A_iou, U1_iou, U2_iou, W_iou, b_ia, b_iou, ws_iou);

  treelstm_epilogue<<<dim3((NN * HH) / 256), 256, 0, stream>>>(
      ws_f, ws_iou, nc, (float*)d_out);
}